// SpikingHabenula_88149908783724
// MI455X (gfx1250) — compile-verified
//
#include <hip/hip_runtime.h>

#define STEPS 20

// CDNA5 async DMA: global -> LDS, 128-bit per lane, tracked by ASYNCcnt.
__device__ __forceinline__ void async_ld_b128(unsigned lds_off, const float* g) {
    asm volatile("global_load_async_to_lds_b128 %0, %1, off"
                 :: "v"(lds_off), "v"(g)
                 : "memory");
}

__device__ __forceinline__ void izh_step(float& V, float& U, float& R, float& ACC,
                                         float nz, float ib) {
    float I  = ib + nz * 0.5f;
    float v2 = V + (0.04f * V * V + 5.0f * V + 140.0f - U + I);
    float u2 = U + 0.02f * (0.2f * v2 - U);
    bool fired = v2 >= 30.0f;
    float spk = fired ? 1.0f : 0.0f;
    V = fired ? -65.0f : v2;
    U = u2 + spk * 8.0f;
    R = R * 0.95f + spk * 0.05f;
    ACC += spk;
}

__global__ __launch_bounds__(256) void izh_pop_kernel(
    const float* __restrict__ v_in, const float* __restrict__ u_in,
    const float* __restrict__ r_in, const float* __restrict__ noise,
    int n, int pop,
    const float* __restrict__ reward, const float* __restrict__ expected,
    const float* __restrict__ aversion,
    float* __restrict__ out_spk, float* __restrict__ out_rate,
    float* __restrict__ partial)
{
    __shared__ float4 stage[2][256];   // double-buffered async-DMA landing zone
    __shared__ float  red[256];

    const int  tid    = threadIdx.x;
    const int  nvec   = n >> 2;                       // n is a multiple of 4
    const int  vi     = blockIdx.x * 256 + tid;
    const bool active = vi < nvec;
    const size_t base = active ? ((size_t)vi << 2) : 0;  // clamp tail lanes to safe addr

    // Tonic current from device-resident scalars (uniform -> scalar loads).
    float ib;
    if (pop == 0) {
        float rpe = *reward - *expected;
        ib = fmaxf(0.0f, -rpe) * 20.0f - 1.0f;        // disappoint*20 + I_TONIC_LHB
    } else if (pop == 1) {
        ib = (*aversion) * 18.0f - 0.5f;              // I_TONIC_RMHB
    } else {
        ib = (*aversion) * 10.0f - 1.5f;              // I_TONIC_LMHB
    }

    float4 V   = *(const float4*)(v_in + base);
    float4 U   = *(const float4*)(u_in + base);
    float4 R   = *(const float4*)(r_in + base);
    float4 ACC = make_float4(0.f, 0.f, 0.f, 0.f);

    unsigned lds[2];
    lds[0] = (unsigned)(size_t)&stage[0][tid];
    lds[1] = (unsigned)(size_t)&stage[1][tid];

    // Prologue: put noise rows 0 and 1 in flight (depth-2 pipeline).
    async_ld_b128(lds[0], noise + base);
    async_ld_b128(lds[1], noise + (size_t)n + base);

    #pragma unroll
    for (int s = 0; s < STEPS; ++s) {
        // Wait until this step's DMA landed (<=1 outstanding, except last step).
        if (s < STEPS - 1)
            asm volatile("s_wait_asynccnt 0x1" ::: "memory");
        else
            asm volatile("s_wait_asynccnt 0x0" ::: "memory");
        float4 nz = stage[s & 1][tid];
        // Ensure the ds_load completed into VGPRs before the async engine may
        // overwrite this buffer (async LDS writes are unordered vs DS reads).
        asm volatile("s_wait_dscnt 0x0" ::: "memory");
        if (s + 2 < STEPS)
            async_ld_b128(lds[s & 1], noise + (size_t)(s + 2) * (size_t)n + base);

        izh_step(V.x, U.x, R.x, ACC.x, nz.x, ib);
        izh_step(V.y, U.y, R.y, ACC.y, nz.y, ib);
        izh_step(V.z, U.z, R.z, ACC.z, nz.z, ib);
        izh_step(V.w, U.w, R.w, ACC.w, nz.w, ib);
    }

    if (active) {
        if (pop == 0) *(float4*)(out_spk + base) = ACC;
        *(float4*)(out_rate + base) = R;
    }

    // Deterministic per-block rate sum -> partials in workspace.
    red[tid] = active ? (R.x + R.y + R.z + R.w) : 0.0f;
    __syncthreads();
    for (int off = 128; off > 0; off >>= 1) {
        if (tid < off) red[tid] += red[tid + off];
        __syncthreads();
    }
    if (tid == 0) partial[blockIdx.x] = red[0];
}

__global__ __launch_bounds__(256) void finalize_kernel(
    const float* __restrict__ ws, int c_lhb, int c_rmhb, int c_lmhb,
    int n_lhb, int n_rmhb, int n_lmhb,
    const float* __restrict__ frustration, const float* __restrict__ reward,
    const float* __restrict__ expected, const float* __restrict__ aversion,
    const float* __restrict__ DA, const int* __restrict__ goal,
    float* __restrict__ out_tail)
{
    __shared__ float red[256];
    __shared__ float sums[3];
    const int tid = threadIdx.x;

    const float* arrs[3] = { ws, ws + c_lhb, ws + c_lhb + c_rmhb };
    const int    cnts[3] = { c_lhb, c_rmhb, c_lmhb };

    for (int k = 0; k < 3; ++k) {
        float s = 0.0f;
        for (int i = tid; i < cnts[k]; i += 256) s += arrs[k][i];  // fixed order
        red[tid] = s;
        __syncthreads();
        for (int off = 128; off > 0; off >>= 1) {
            if (tid < off) red[tid] += red[tid + off];
            __syncthreads();
        }
        if (tid == 0) sums[k] = red[0];
        __syncthreads();
    }

    if (tid == 0) {
        float lhb_mean  = sums[0] / (float)n_lhb;
        float rmhb_mean = sums[1] / (float)n_rmhb;
        float lmhb_mean = sums[2] / (float)n_lmhb;
        float mhb_mean  = 0.67f * rmhb_mean + 0.33f * lmhb_mean;

        float rpe = *reward - *expected;
        float dis = fmaxf(0.0f, -rpe);
        float da_supp = fminf(0.5f, lhb_mean * 5.0f);
        float ht5     = fminf(0.3f, lhb_mean * 3.0f);
        float ach     = fminf(1.0f, rmhb_mean * 4.0f);
        float expl    = fminf(1.0f, dis * 2.0f + lhb_mean * 3.0f);

        int g = *goal;
        float fr[4];
        #pragma unroll
        for (int k = 0; k < 4; ++k) fr[k] = frustration[k] * 0.998f;
        if (rpe < -0.05f) fr[g] += 0.05f * fabsf(rpe);
        if (rpe >  0.1f)  fr[g] *= 0.8f;
        #pragma unroll
        for (int k = 0; k < 4; ++k) fr[k] = fminf(1.0f, fmaxf(0.0f, fr[k]));
        float helpless = fmaxf(fmaxf(fr[0], fr[1]), fmaxf(fr[2], fr[3]));
        float sw = (fr[g] > 0.4f) ? 1.0f : 0.0f;
        if (sw > 0.0f) fr[g] *= 0.3f;
        float dopa = fmaxf(0.5f, 2.0f * (1.0f - *DA));

        out_tail[0]  = dis;       out_tail[1] = da_supp;  out_tail[2] = ht5;
        out_tail[3]  = lhb_mean;  out_tail[4] = mhb_mean; out_tail[5] = rmhb_mean;
        out_tail[6]  = lmhb_mean; out_tail[7] = ach;      out_tail[8] = expl;
        out_tail[9]  = sw;        out_tail[10] = helpless;
        #pragma unroll
        for (int k = 0; k < 4; ++k) out_tail[11 + k] = fr[k] * 0.5f * dopa;
    }
}

extern "C" void kernel_launch(void* const* d_in, const int* in_sizes, int n_in,
                              void* d_out, int out_size, void* d_ws, size_t ws_size,
                              hipStream_t stream) {
    (void)n_in; (void)out_size; (void)ws_size;
    const float* v_lhb  = (const float*)d_in[0];
    const float* u_lhb  = (const float*)d_in[1];
    const float* r_lhb  = (const float*)d_in[2];
    const float* v_rm   = (const float*)d_in[3];
    const float* u_rm   = (const float*)d_in[4];
    const float* r_rm   = (const float*)d_in[5];
    const float* v_lm   = (const float*)d_in[6];
    const float* u_lm   = (const float*)d_in[7];
    const float* r_lm   = (const float*)d_in[8];
    const float* nz_lhb = (const float*)d_in[9];
    const float* nz_rm  = (const float*)d_in[10];
    const float* nz_lm  = (const float*)d_in[11];
    const float* frus   = (const float*)d_in[12];
    const float* reward = (const float*)d_in[13];
    const float* expect = (const float*)d_in[14];
    const float* avers  = (const float*)d_in[15];
    const float* DA     = (const float*)d_in[16];
    const int*   goal   = (const int*)d_in[17];

    const int n_lhb = in_sizes[0];
    const int n_rm  = in_sizes[3];
    const int n_lm  = in_sizes[6];

    float* out = (float*)d_out;
    float* ws  = (float*)d_ws;

    const int b_lhb = (n_lhb / 4 + 255) / 256;
    const int b_rm  = (n_rm  / 4 + 255) / 256;
    const int b_lm  = (n_lm  / 4 + 255) / 256;

    float* p_lhb = ws;
    float* p_rm  = ws + b_lhb;
    float* p_lm  = p_rm + b_rm;

    float* o_spk  = out;
    float* o_rlhb = out + (size_t)n_lhb;
    float* o_rrm  = out + 2 * (size_t)n_lhb;
    float* o_rlm  = o_rrm + n_rm;
    float* o_tail = o_rlm + n_lm;

    izh_pop_kernel<<<b_lhb, 256, 0, stream>>>(v_lhb, u_lhb, r_lhb, nz_lhb, n_lhb, 0,
                                              reward, expect, avers, o_spk, o_rlhb, p_lhb);
    izh_pop_kernel<<<b_rm, 256, 0, stream>>>(v_rm, u_rm, r_rm, nz_rm, n_rm, 1,
                                             reward, expect, avers, nullptr, o_rrm, p_rm);
    izh_pop_kernel<<<b_lm, 256, 0, stream>>>(v_lm, u_lm, r_lm, nz_lm, n_lm, 2,
                                             reward, expect, avers, nullptr, o_rlm, p_lm);
    finalize_kernel<<<1, 256, 0, stream>>>(ws, b_lhb, b_rm, b_lm, n_lhb, n_rm, n_lm,
                                           frus, reward, expect, avers, DA, goal, o_tail);
}